// RoiPoolingConv_498216206531
// MI455X (gfx1250) — compile-verified
//
#include <hip/hip_runtime.h>
#include <cstdint>
#include <cstddef>

// ROI bilinear pooling for MI455X (gfx1250).
// img: (1,64,64,1024) f32 ; rois: (1,300,4) i32 (x,y,w,h) ; out: (1,300,14,14,1024) f32
//
// Memory-bound problem: 241 MB of NT stores (~10.3 us @ 23.3 TB/s) dominates;
// the 16.8 MB image lives in the 192 MB L2. CDNA5 path used: async global->LDS
// DMA (ASYNCcnt) with double buffering to hide gather latency, no barriers
// (each thread consumes only the LDS words it requested).

#define POOL    14
#define IMG_W   64
#define CHANS   1024
#define THREADS 256

typedef float vfloat4 __attribute__((ext_vector_type(4)));

// Match the builtin's parameter types exactly: GCC-style int vector(4),
// AS1 (global) for the source, AS3 (LDS) for the destination.
typedef int v4i_vs __attribute__((vector_size(16)));
typedef __attribute__((address_space(1))) v4i_vs* g_v4i_ptr;
typedef __attribute__((address_space(3))) v4i_vs* l_v4i_ptr;

#if defined(__HIP_DEVICE_COMPILE__) && __has_builtin(__builtin_amdgcn_global_load_async_to_lds_b128)
#define HAVE_ASYNC_LDS 1
#else
#define HAVE_ASYNC_LDS 0
#endif

#if defined(__HIP_DEVICE_COMPILE__) && __has_builtin(__builtin_amdgcn_s_wait_asynccnt)
#define WAIT_ASYNC(n) __builtin_amdgcn_s_wait_asynccnt(n)
#elif defined(__HIP_DEVICE_COMPILE__)
#define WAIT_ASYNC(n) asm volatile("s_wait_asynccnt %0" ::"i"(n) : "memory")
#else
#define WAIT_ASYNC(n)
#endif

static __device__ __forceinline__ int imin_i(int a, int b) { return a < b ? a : b; }

__global__ __launch_bounds__(THREADS) void roi_pool_kernel(
    const float* __restrict__ img, const int* __restrict__ rois,
    float* __restrict__ out)
{
    const int blk = blockIdx.x;          // one block per (roi, py)
    const int r   = blk / POOL;
    const int py  = blk - r * POOL;
    const int t   = threadIdx.x;         // 0..255
    const int ch0 = t * 4;               // 4 channels per thread (256*4 = 1024)

    // Block-uniform ROI fields -> scalar loads.
    const int rx = rois[r * 4 + 0];
    const int ry = rois[r * 4 + 1];
    const int rw = rois[r * 4 + 2];
    const int rh = rois[r * 4 + 3];

    // TF1 legacy resize: src = out_idx * (in/out), no half-pixel offset.
    const float sy = (float)rh / (float)POOL;
    const float sx = (float)rw / (float)POOL;
    const float fy = (float)py * sy;
    const int   y0 = (int)fy;                      // fy >= 0
    const int   y1 = imin_i(y0 + 1, rh - 1);
    const float ly = fy - (float)y0;

    const float* row0 = img + (size_t)(ry + y0) * IMG_W * CHANS + ch0;
    const float* row1 = img + (size_t)(ry + y1) * IMG_W * CHANS + ch0;
    float*       obase = out + (size_t)(r * POOL + py) * POOL * CHANS + ch0;

#if HAVE_ASYNC_LDS
    // Double buffer: [buf][tap][channel]; each stage DMAs 4 taps x 4KB.
    __shared__ __attribute__((aligned(16))) float lds[2][4][CHANS];

    auto issue = [&](int px, int buf) {
        const float fx = (float)px * sx;
        const int   x0 = (int)fx;
        const int   x1 = imin_i(x0 + 1, rw - 1);
        const size_t c0 = (size_t)(rx + x0) * CHANS;
        const size_t c1 = (size_t)(rx + x1) * CHANS;
        __builtin_amdgcn_global_load_async_to_lds_b128(
            (g_v4i_ptr)(void*)(row0 + c0), (l_v4i_ptr)(void*)&lds[buf][0][ch0], 0, 0);
        __builtin_amdgcn_global_load_async_to_lds_b128(
            (g_v4i_ptr)(void*)(row0 + c1), (l_v4i_ptr)(void*)&lds[buf][1][ch0], 0, 0);
        __builtin_amdgcn_global_load_async_to_lds_b128(
            (g_v4i_ptr)(void*)(row1 + c0), (l_v4i_ptr)(void*)&lds[buf][2][ch0], 0, 0);
        __builtin_amdgcn_global_load_async_to_lds_b128(
            (g_v4i_ptr)(void*)(row1 + c1), (l_v4i_ptr)(void*)&lds[buf][3][ch0], 0, 0);
    };

    auto compute = [&](int px, int buf) {
        const float fx = (float)px * sx;
        const int   x0 = (int)fx;
        const float lx = fx - (float)x0;
        vfloat4 v00 = *(const vfloat4*)&lds[buf][0][ch0];
        vfloat4 v01 = *(const vfloat4*)&lds[buf][1][ch0];
        vfloat4 v10 = *(const vfloat4*)&lds[buf][2][ch0];
        vfloat4 v11 = *(const vfloat4*)&lds[buf][3][ch0];
        vfloat4 top = v00 + (v01 - v00) * lx;
        vfloat4 bot = v10 + (v11 - v10) * lx;
        vfloat4 res = top + (bot - top) * ly;
        __builtin_nontemporal_store(res, (vfloat4*)(obase + (size_t)px * CHANS));
    };

    issue(0, 0);
    for (int px = 0; px < POOL; ++px) {
        const int buf = px & 1;
        if (px + 1 < POOL) {
            issue(px + 1, buf ^ 1);
            WAIT_ASYNC(4);   // 4 async instrs/wave per stage; in-order retire
        } else {
            WAIT_ASYNC(0);
        }
        asm volatile("" ::: "memory");  // don't hoist LDS reads above the wait
        compute(px, buf);
    }
#else
    // Fallback: direct gather through L2 (image is L2-resident anyway).
    for (int px = 0; px < POOL; ++px) {
        const float fx = (float)px * sx;
        const int   x0 = (int)fx;
        const int   x1 = imin_i(x0 + 1, rw - 1);
        const float lx = fx - (float)x0;
        const size_t c0 = (size_t)(rx + x0) * CHANS;
        const size_t c1 = (size_t)(rx + x1) * CHANS;
        vfloat4 v00 = *(const vfloat4*)(row0 + c0);
        vfloat4 v01 = *(const vfloat4*)(row0 + c1);
        vfloat4 v10 = *(const vfloat4*)(row1 + c0);
        vfloat4 v11 = *(const vfloat4*)(row1 + c1);
        vfloat4 top = v00 + (v01 - v00) * lx;
        vfloat4 bot = v10 + (v11 - v10) * lx;
        vfloat4 res = top + (bot - top) * ly;
        __builtin_nontemporal_store(res, (vfloat4*)(obase + (size_t)px * CHANS));
    }
#endif
}

extern "C" void kernel_launch(void* const* d_in, const int* in_sizes, int n_in,
                              void* d_out, int out_size, void* d_ws, size_t ws_size,
                              hipStream_t stream) {
    (void)n_in; (void)out_size; (void)d_ws; (void)ws_size;
    const float* img  = (const float*)d_in[0];
    const int*   rois = (const int*)d_in[1];
    float*       out  = (float*)d_out;

    const int R = in_sizes[1] / 4;       // 300 ROIs
    dim3 grid(R * POOL);
    dim3 block(THREADS);
    hipLaunchKernelGGL(roi_pool_kernel, grid, block, 0, stream, img, rois, out);
}